// SpatiallyModulatedAttention_67276367725149
// MI455X (gfx1250) — compile-verified
//
#include <hip/hip_runtime.h>
#include <hip/hip_bf16.h>

// ---------------- problem constants ----------------
#define DMODEL 1024
#define NHEAD  16
#define DK     64
#define SEQL   2048
#define BATCH  2

typedef float v2f __attribute__((ext_vector_type(2)));
typedef float v8f __attribute__((ext_vector_type(8)));

// CDNA5 fp32 matrix core: D(16x16,f32) = A(16x4,f32) x B(4x16,f32) + C
__device__ __forceinline__ v8f wmma4(v2f a, v2f b, v8f c) {
    return __builtin_amdgcn_wmma_f32_16x16x4_f32(
        /*neg_a=*/false, a, /*neg_b=*/false, b,
        /*c_mod=*/(short)0, c, /*reuse_a=*/false, /*reuse_b=*/false);
}

__device__ __forceinline__ v2f ntload2(const float* p) {
    return __builtin_nontemporal_load((const v2f*)p);
}

// ---------------------------------------------------------------------------
// Software-pipelined 16x64-slab GEMM body: y[m0:m0+16, n0:n0+64] over K kdim.
// A rows from `arow` (stride 1 in k), B rows from br0..br3 (stride 1 in k).
// All pointers pre-offset by the lane's kb. Results in a0..a3.
// ---------------------------------------------------------------------------
#define GEMM_SLAB_LOOP(KDIM, AR, B0, B1, B2, B3)                         \
    v2f a_c  = *(const v2f*)(AR);                                        \
    v2f b0_c = *(const v2f*)(B0);                                        \
    v2f b1_c = *(const v2f*)(B1);                                        \
    v2f b2_c = *(const v2f*)(B2);                                        \
    v2f b3_c = *(const v2f*)(B3);                                        \
    _Pragma("unroll 4")                                                  \
    for (int kk = 0; kk < (KDIM) - 4; kk += 4) {                         \
        v2f a_n  = *(const v2f*)((AR) + kk + 4);                         \
        v2f b0_n = *(const v2f*)((B0) + kk + 4);                         \
        v2f b1_n = *(const v2f*)((B1) + kk + 4);                         \
        v2f b2_n = *(const v2f*)((B2) + kk + 4);                         \
        v2f b3_n = *(const v2f*)((B3) + kk + 4);                         \
        a0 = wmma4(a_c, b0_c, a0);                                       \
        a1 = wmma4(a_c, b1_c, a1);                                       \
        a2 = wmma4(a_c, b2_c, a2);                                       \
        a3 = wmma4(a_c, b3_c, a3);                                       \
        a_c = a_n; b0_c = b0_n; b1_c = b1_n; b2_c = b2_n; b3_c = b3_n;   \
    }                                                                    \
    a0 = wmma4(a_c, b0_c, a0);                                           \
    a1 = wmma4(a_c, b1_c, a1);                                           \
    a2 = wmma4(a_c, b2_c, a2);                                           \
    a3 = wmma4(a_c, b3_c, a3);

// ---------------------------------------------------------------------------
// K1: fused Q/K/V projections. y = x @ W.T, x taken from [L,B,D] input.
// Each wave computes a 16x64 slab (one head). grid: (B*L/128, D/64, 3), blk 256
// ---------------------------------------------------------------------------
__global__ void proj_kernel(const float* __restrict__ q, const float* __restrict__ k,
                            const float* __restrict__ v,
                            const float* __restrict__ Wq, const float* __restrict__ Wk,
                            const float* __restrict__ Wv,
                            float* __restrict__ qh, float* __restrict__ kh,
                            float* __restrict__ vh) {
    const float* x; const float* W; float* dst;
    if (blockIdx.z == 0)      { x = q; W = Wq; dst = qh; }
    else if (blockIdx.z == 1) { x = k; W = Wk; dst = kh; }
    else                      { x = v; W = Wv; dst = vh; }

    const int wave = threadIdx.x >> 5;
    const int lane = threadIdx.x & 31;
    const int hp   = lane >> 4;            // half-wave (selects K pair / row offset)
    const int l16  = lane & 15;
    const int kb   = hp * 2;

    const int m0 = blockIdx.x * 128 + wave * 16;   // row tile in [B*L]
    const int n0 = blockIdx.y * 64;                // output-feature slab (one head)

    const int ia = m0 + l16;                       // A-fragment row
    const int ba = ia >> 11, la = ia & (SEQL - 1); // row -> (b,l)
    const float* ar = x + ((size_t)(la * BATCH + ba)) * DMODEL + kb; // x[l][b][*]
    const float* b0 = W + ((size_t)(n0      + l16)) * DMODEL + kb;   // W[n][*]
    const float* b1 = W + ((size_t)(n0 + 16 + l16)) * DMODEL + kb;
    const float* b2 = W + ((size_t)(n0 + 32 + l16)) * DMODEL + kb;
    const float* b3 = W + ((size_t)(n0 + 48 + l16)) * DMODEL + kb;

    v8f a0 = {}, a1 = {}, a2 = {}, a3 = {};
    GEMM_SLAB_LOOP(DMODEL, ar, b0, b1, b2, b3)

    const int h = n0 >> 6;                 // whole slab lies inside one head
    #pragma unroll
    for (int r = 0; r < 8; ++r) {
        const int i = m0 + r + 8 * hp;
        const int b_ = i >> 11, l_ = i & (SEQL - 1);
        float* drow = dst + (((size_t)(b_ * NHEAD + h)) * SEQL + l_) * DK + l16;
        drow[0]  = a0[r];
        drow[16] = a1[r];
        drow[32] = a2[r];
        drow[48] = a3[r];
    }
}

// ---------------------------------------------------------------------------
// K2: scores = (qh @ kh.T)/8 * sph, masked -> written into p_attn region.
// One wave: 16 q-rows x 64 j-cols. sph/mask single-use -> non-temporal loads.
// grid: (L/128, L/64, B*H), block 256
// ---------------------------------------------------------------------------
__global__ void scores_kernel(const float* __restrict__ qh, const float* __restrict__ kh,
                              const float* __restrict__ sph, const int* __restrict__ mask,
                              float* __restrict__ pa) {
    const int z = blockIdx.z;          // b*H + h
    const int b = z >> 4;
    const int wave = threadIdx.x >> 5, lane = threadIdx.x & 31;
    const int hp = lane >> 4, l16 = lane & 15, kb = hp * 2;

    const int q0 = (blockIdx.x * 8 + wave) * 16;
    const int j0 = blockIdx.y * 64;

    const float* ar = qh + ((size_t)z * SEQL + (q0 + l16)) * DK + kb;
    const float* b0 = kh + ((size_t)z * SEQL + (j0      + l16)) * DK + kb;
    const float* b1 = kh + ((size_t)z * SEQL + (j0 + 16 + l16)) * DK + kb;
    const float* b2 = kh + ((size_t)z * SEQL + (j0 + 32 + l16)) * DK + kb;
    const float* b3 = kh + ((size_t)z * SEQL + (j0 + 48 + l16)) * DK + kb;

    v8f a0 = {}, a1 = {}, a2 = {}, a3 = {};
    GEMM_SLAB_LOOP(DK, ar, b0, b1, b2, b3)

    #pragma unroll
    for (int r = 0; r < 8; ++r) {
        const int qi = q0 + r + 8 * hp;
        const size_t rbase = ((size_t)z * SEQL + qi) * SEQL + j0 + l16;
        const size_t mbase = ((size_t)b * SEQL + qi) * SEQL + j0 + l16;
        const float s0 = a0[r], s1 = a1[r], s2 = a2[r], s3 = a3[r];
        const float p0 = __builtin_nontemporal_load(sph + rbase);
        const float p1 = __builtin_nontemporal_load(sph + rbase + 16);
        const float p2 = __builtin_nontemporal_load(sph + rbase + 32);
        const float p3 = __builtin_nontemporal_load(sph + rbase + 48);
        const int   m0v = __builtin_nontemporal_load(mask + mbase);
        const int   m1v = __builtin_nontemporal_load(mask + mbase + 16);
        const int   m2v = __builtin_nontemporal_load(mask + mbase + 32);
        const int   m3v = __builtin_nontemporal_load(mask + mbase + 48);
        pa[rbase]      = (m0v == 0) ? -1e9f : s0 * 0.125f * p0;
        pa[rbase + 16] = (m1v == 0) ? -1e9f : s1 * 0.125f * p1;
        pa[rbase + 32] = (m2v == 0) ? -1e9f : s2 * 0.125f * p2;
        pa[rbase + 48] = (m3v == 0) ? -1e9f : s3 * 0.125f * p3;
    }
}

// ---------------------------------------------------------------------------
// K3: row softmax over last dim (2048), in place. grid: B*H*L blocks of 256.
// ---------------------------------------------------------------------------
__global__ void softmax_kernel(float* __restrict__ pa) {
    float* r = pa + (size_t)blockIdx.x * SEQL;
    __shared__ float red[8];
    const int t = threadIdx.x, lane = t & 31, w = t >> 5;

    float v[8];
    float mx = -3.4e38f;
    #pragma unroll
    for (int i = 0; i < 8; ++i) { v[i] = r[t + i * 256]; mx = fmaxf(mx, v[i]); }
    #pragma unroll
    for (int off = 16; off; off >>= 1) mx = fmaxf(mx, __shfl_xor(mx, off));
    if (lane == 0) red[w] = mx;
    __syncthreads();
    mx = red[0];
    #pragma unroll
    for (int i = 1; i < 8; ++i) mx = fmaxf(mx, red[i]);

    float s = 0.f;
    #pragma unroll
    for (int i = 0; i < 8; ++i) { v[i] = __expf(v[i] - mx); s += v[i]; }
    #pragma unroll
    for (int off = 16; off; off >>= 1) s += __shfl_xor(s, off);
    __syncthreads();
    if (lane == 0) red[w] = s;
    __syncthreads();
    s = 0.f;
    #pragma unroll
    for (int i = 0; i < 8; ++i) s += red[i];

    const float inv = 1.0f / s;
    #pragma unroll
    for (int i = 0; i < 8; ++i) r[t + i * 256] = v[i] * inv;
}

// ---------------------------------------------------------------------------
// K4: out = p_attn @ vh  -> attn_out[b][l][h*64+dk]  ([B,L,D] merged-head)
// One wave computes 16 q-rows x all 64 d-cols: p_attn (the 537MB stream) is
// read exactly once, non-temporally; vh (512KB per bh) stays cache-resident.
// Software-pipelined. grid: (L/128, 1, B*H), block 256
// ---------------------------------------------------------------------------
__global__ void pv_kernel(const float* __restrict__ pa, const float* __restrict__ vh,
                          float* __restrict__ ao) {
    const int z = blockIdx.z, b = z >> 4, h = z & 15;
    const int wave = threadIdx.x >> 5, lane = threadIdx.x & 31;
    const int hp = lane >> 4, l16 = lane & 15, kb = hp * 2;

    const int q0 = (blockIdx.x * 8 + wave) * 16;

    const float* ar = pa + ((size_t)z * SEQL + (q0 + l16)) * SEQL + kb;
    const float* vb = vh + (size_t)z * SEQL * DK + (size_t)kb * DK + l16;

    v8f a0 = {}, a1 = {}, a2 = {}, a3 = {};

    v2f a_c = ntload2(ar);
    v2f b0_c = { vb[0],  vb[DK]      };
    v2f b1_c = { vb[16], vb[DK + 16] };
    v2f b2_c = { vb[32], vb[DK + 32] };
    v2f b3_c = { vb[48], vb[DK + 48] };
    #pragma unroll 2
    for (int kk = 0; kk < SEQL - 4; kk += 4) {
        const float* vn = vb + (size_t)(kk + 4) * DK;
        v2f a_n  = ntload2(ar + kk + 4);
        v2f b0_n = { vn[0],  vn[DK]      };
        v2f b1_n = { vn[16], vn[DK + 16] };
        v2f b2_n = { vn[32], vn[DK + 32] };
        v2f b3_n = { vn[48], vn[DK + 48] };
        a0 = wmma4(a_c, b0_c, a0);
        a1 = wmma4(a_c, b1_c, a1);
        a2 = wmma4(a_c, b2_c, a2);
        a3 = wmma4(a_c, b3_c, a3);
        a_c = a_n; b0_c = b0_n; b1_c = b1_n; b2_c = b2_n; b3_c = b3_n;
    }
    a0 = wmma4(a_c, b0_c, a0);
    a1 = wmma4(a_c, b1_c, a1);
    a2 = wmma4(a_c, b2_c, a2);
    a3 = wmma4(a_c, b3_c, a3);

    #pragma unroll
    for (int r = 0; r < 8; ++r) {
        const int qi = q0 + r + 8 * hp;
        float* drow = ao + ((size_t)b * SEQL + qi) * DMODEL + h * DK + l16;
        drow[0]  = a0[r];
        drow[16] = a1[r];
        drow[32] = a2[r];
        drow[48] = a3[r];
    }
}

// ---------------------------------------------------------------------------
// K5: fc_out = attn_out @ Wfc.T + residual(x_q). 16x64 slab per wave.
// grid: (B*L/128, DMODEL/64), block 256
// ---------------------------------------------------------------------------
__global__ void fc_kernel(const float* __restrict__ ao, const float* __restrict__ Wfc,
                          const float* __restrict__ q, float* __restrict__ fo) {
    const int wave = threadIdx.x >> 5, lane = threadIdx.x & 31;
    const int hp = lane >> 4, l16 = lane & 15, kb = hp * 2;

    const int m0 = blockIdx.x * 128 + wave * 16;
    const int n0 = blockIdx.y * 64;

    const float* ar = ao  + (size_t)(m0 + l16) * DMODEL + kb;
    const float* b0 = Wfc + (size_t)(n0      + l16) * DMODEL + kb;
    const float* b1 = Wfc + (size_t)(n0 + 16 + l16) * DMODEL + kb;
    const float* b2 = Wfc + (size_t)(n0 + 32 + l16) * DMODEL + kb;
    const float* b3 = Wfc + (size_t)(n0 + 48 + l16) * DMODEL + kb;

    v8f a0 = {}, a1 = {}, a2 = {}, a3 = {};
    GEMM_SLAB_LOOP(DMODEL, ar, b0, b1, b2, b3)

    #pragma unroll
    for (int r = 0; r < 8; ++r) {
        const int i = m0 + r + 8 * hp;
        const int b_ = i >> 11, l_ = i & (SEQL - 1);
        const float* rrow = q + ((size_t)l_ * BATCH + b_) * DMODEL + n0 + l16;
        float* drow = fo + (size_t)i * DMODEL + n0 + l16;
        drow[0]  = a0[r] + rrow[0];
        drow[16] = a1[r] + rrow[16];
        drow[32] = a2[r] + rrow[32];
        drow[48] = a3[r] + rrow[48];
    }
}

// ---------------------------------------------------------------------------
// K6: LayerNorm (biased var, eps=1e-6) + transpose store to [L,B,D].
// grid: B*L blocks of 256 (4 elems/thread)
// ---------------------------------------------------------------------------
__global__ void ln_kernel(const float* __restrict__ fo, const float* __restrict__ gamma,
                          const float* __restrict__ beta, float* __restrict__ out) {
    const int i = blockIdx.x;
    const int b = i >> 11, l = i & (SEQL - 1);
    const float* r = fo + (size_t)i * DMODEL;
    __shared__ float rs[8], rs2[8];
    const int t = threadIdx.x, lane = t & 31, w = t >> 5;

    float x[4]; float s = 0.f, s2 = 0.f;
    #pragma unroll
    for (int j = 0; j < 4; ++j) {
        x[j] = r[t + j * 256];
        s += x[j]; s2 += x[j] * x[j];
    }
    #pragma unroll
    for (int off = 16; off; off >>= 1) { s += __shfl_xor(s, off); s2 += __shfl_xor(s2, off); }
    if (lane == 0) { rs[w] = s; rs2[w] = s2; }
    __syncthreads();
    s = 0.f; s2 = 0.f;
    #pragma unroll
    for (int j = 0; j < 8; ++j) { s += rs[j]; s2 += rs2[j]; }

    const float mu  = s * (1.0f / DMODEL);
    const float var = s2 * (1.0f / DMODEL) - mu * mu;
    const float sc  = rsqrtf(var + 1e-6f);

    float* orow = out + ((size_t)l * BATCH + b) * DMODEL;
    #pragma unroll
    for (int j = 0; j < 4; ++j) {
        const int c = t + j * 256;
        orow[c] = (x[j] - mu) * sc * gamma[c] + beta[c];
    }
}

// ---------------------------------------------------------------------------
extern "C" void kernel_launch(void* const* d_in, const int* in_sizes, int n_in,
                              void* d_out, int out_size, void* d_ws, size_t ws_size,
                              hipStream_t stream) {
    (void)in_sizes; (void)n_in; (void)out_size; (void)ws_size;

    const float* q    = (const float*)d_in[0];
    const float* k    = (const float*)d_in[1];
    const float* v    = (const float*)d_in[2];
    const float* sph  = (const float*)d_in[3];
    const int*   mask = (const int*)  d_in[4];
    const float* Wq   = (const float*)d_in[5];
    const float* Wk   = (const float*)d_in[6];
    const float* Wv   = (const float*)d_in[7];
    const float* Wfc  = (const float*)d_in[8];
    const float* gamma= (const float*)d_in[9];
    const float* beta = (const float*)d_in[10];

    float* out = (float*)d_out;                              // [L,B,D]
    float* pa  = out + (size_t)SEQL * BATCH * DMODEL;        // p_attn [B,H,L,L]

    const size_t HEADBUF = (size_t)BATCH * NHEAD * SEQL * DK; // 4M floats
    float* ws = (float*)d_ws;
    float* qh = ws;                   // live: K1..K2
    float* kh = ws + HEADBUF;         // live: K1..K2
    float* vh = ws + 2 * HEADBUF;     // live: K1..K4
    float* ao = qh;                   // reuse qh region (free after K2)
    float* fo = kh;                   // reuse kh region (free after K2)

    // 1) projections (3 GEMMs fused via grid.z)
    proj_kernel<<<dim3((BATCH * SEQL) / 128, DMODEL / 64, 3), 256, 0, stream>>>(
        q, k, v, Wq, Wk, Wv, qh, kh, vh);
    // 2) modulated, masked scores -> p_attn region
    scores_kernel<<<dim3(SEQL / 128, SEQL / 64, BATCH * NHEAD), 256, 0, stream>>>(
        qh, kh, sph, mask, pa);
    // 3) softmax rows in place
    softmax_kernel<<<dim3(BATCH * NHEAD * SEQL), 256, 0, stream>>>(pa);
    // 4) p_attn @ V (p_attn streamed once, NT)
    pv_kernel<<<dim3(SEQL / 128, 1, BATCH * NHEAD), 256, 0, stream>>>(pa, vh, ao);
    // 5) fc + residual
    fc_kernel<<<dim3((BATCH * SEQL) / 128, DMODEL / 64), 256, 0, stream>>>(ao, Wfc, q, fo);
    // 6) layernorm + transpose to [L,B,D]
    ln_kernel<<<dim3(BATCH * SEQL), 256, 0, stream>>>(fo, gamma, beta, out);
}